// LinPositionalAttention_7559142441216
// MI455X (gfx1250) — compile-verified
//
#include <hip/hip_runtime.h>
#include <cstdint>

#define N_B 2
#define S_LEN 1024
#define H_N 12
#define D_DIM 64
#define HIDDEN (H_N * D_DIM)
#define EPS_F 1e-6f

typedef __attribute__((ext_vector_type(2))) float v2f;
typedef __attribute__((ext_vector_type(4))) float f4;
typedef __attribute__((ext_vector_type(8))) float v8f;

__device__ __forceinline__ float elu1(float x) {
    // elu(x) + 1 : x > 0 ? x + 1 : exp(x)
    return x > 0.0f ? x + 1.0f : __expf(x);
}

// ---------------------------------------------------------------------------
// Stage 1: pv[n,h,m,d] = sum_s v[n,s,h,m] * p[n,s,h,d], then *= head_mask[h]
// GEMM per (n,h): C(64x64) = A(64x1024) * B(1024x64), A = v^T, B = p.
// One wave per 16x16 tile using V_WMMA_F32_16X16X4_F32 (fp32 WMMA).
// grid = N*H*16 blocks of 32 threads (1 wave each).
// ---------------------------------------------------------------------------
__global__ void pv_wmma_kernel(const float* __restrict__ v,
                               const float* __restrict__ pos_weight,
                               const float* __restrict__ amask,
                               const float* __restrict__ hmask,
                               float* __restrict__ pv) {
    const int b    = blockIdx.x;
    const int tile = b & 15;
    const int nh   = b >> 4;
    const int n    = nh / H_N;
    const int h    = nh % H_N;
    const int m0   = (tile >> 2) * 16;   // output row tile (m)
    const int d0   = (tile & 3) * 16;    // output col tile (d)

    const int lane  = threadIdx.x;       // wave32
    const int r     = lane & 15;         // A: row-in-tile, B: col-in-tile
    const int khalf = lane >> 4;         // 0 -> K{0,1}, 1 -> K{2,3}

    v8f acc = {};

    const float* vbase  = v + (size_t)n * S_LEN * HIDDEN + (size_t)h * D_DIM;
    const float* pwbase = pos_weight + (size_t)h * D_DIM;
    const float* am     = amask + (size_t)n * S_LEN;

    for (int k = 0; k < S_LEN; k += 4) {
        const int s0 = k + 2 * khalf;
        const int s1 = s0 + 1;

        // A[m, s] = v[n, s, h, m]   (16x4 fp32 tile, 2 VGPRs/lane)
        v2f a;
        a.x = vbase[(size_t)s0 * HIDDEN + (m0 + r)];
        a.y = vbase[(size_t)s1 * HIDDEN + (m0 + r)];

        // B[s, d] = p[n, s, h, d]   (4x16 fp32 tile, 2 VGPRs/lane)
        v2f bv;
        bv.x = elu1(pwbase[(size_t)s0 * HIDDEN + (d0 + r)]) * am[s0];
        bv.y = elu1(pwbase[(size_t)s1 * HIDDEN + (d0 + r)]) * am[s1];

        acc = __builtin_amdgcn_wmma_f32_16x16x4_f32(
            /*neg_a=*/false, a, /*neg_b=*/false, bv,
            /*c_mod=*/(short)0, acc, /*reuse_a=*/false, /*reuse_b=*/false);
    }

    const float hm = hmask[h];
    float* out = pv + (((size_t)n * H_N + h) * D_DIM) * D_DIM;
    // C/D layout: VGPR i -> M = m0 + i + 8*khalf, N = d0 + r
    #pragma unroll
    for (int i = 0; i < 8; ++i) {
        const int m = m0 + i + 8 * khalf;
        out[(size_t)m * D_DIM + (d0 + r)] = acc[i] * hm;
    }
}

// ---------------------------------------------------------------------------
// Stage 2: psum[n,h,d] = sum_l p[n,l,h,d].  grid = N*H blocks of 64 threads.
// ---------------------------------------------------------------------------
__global__ void psum_kernel(const float* __restrict__ pos_weight,
                            const float* __restrict__ amask,
                            float* __restrict__ psum) {
    const int nh = blockIdx.x;
    const int n  = nh / H_N;
    const int h  = nh % H_N;
    const int d  = threadIdx.x;

    const float* pw = pos_weight + (size_t)h * D_DIM + d;
    const float* am = amask + (size_t)n * S_LEN;

    float acc = 0.0f;
    for (int l = 0; l < S_LEN; ++l)
        acc += elu1(pw[(size_t)l * HIDDEN]) * am[l];
    psum[(size_t)nh * D_DIM + d] = acc;
}

// ---------------------------------------------------------------------------
// Stage 3 (the 402.6 MB one): ppv[n,l,h,m,d] = p[n,l,h,d] * pv[n,h,m,d].
// One block per (n,l,h); 256 threads each emit 4 float4 NT stores (16 KB/blk).
// Coalesced: consecutive lanes cover consecutive float4's along d.
// ---------------------------------------------------------------------------
__global__ void ppv_kernel(const float* __restrict__ pos_weight,
                           const float* __restrict__ amask,
                           const float* __restrict__ pv,
                           float* __restrict__ out) {
    const int b = blockIdx.x;            // ordered (n, l, h)
    const int h = b % H_N;
    const int l = (b / H_N) % S_LEN;
    const int n = b / (H_N * S_LEN);

    const int t    = threadIdx.x;        // 0..255
    const int d4   = t & 15;             // float4 index along d
    const int mrow = t >> 4;             // 0..15

    const float m = amask[(size_t)n * S_LEN + l];
    const float* pw = pos_weight + (size_t)l * HIDDEN + (size_t)h * D_DIM + d4 * 4;

    f4 p4;
    p4.x = elu1(pw[0]) * m;
    p4.y = elu1(pw[1]) * m;
    p4.z = elu1(pw[2]) * m;
    p4.w = elu1(pw[3]) * m;

    const f4* pvt = (const f4*)(pv + (((size_t)n * H_N + h) * D_DIM) * D_DIM);
    f4* o = (f4*)(out + (size_t)b * (D_DIM * D_DIM));

    #pragma unroll
    for (int i = 0; i < 4; ++i) {
        const int mm = mrow + 16 * i;
        const f4 q = pvt[mm * 16 + d4];
        f4 res;
        res.x = p4.x * q.x;
        res.y = p4.y * q.y;
        res.z = p4.z * q.z;
        res.w = p4.w * q.w;
        __builtin_nontemporal_store(res, &o[mm * 16 + d4]);   // NT b128 store
    }
}

// ---------------------------------------------------------------------------
// Stage 4: z_pp[n,l,h] = sum_d p[n,l,h,d] * psum[n,h,d] + EPS
// ---------------------------------------------------------------------------
__global__ void z_kernel(const float* __restrict__ pos_weight,
                         const float* __restrict__ amask,
                         const float* __restrict__ psum,
                         float* __restrict__ z) {
    const int g = blockIdx.x * blockDim.x + threadIdx.x;
    if (g >= N_B * S_LEN * H_N) return;
    const int h = g % H_N;
    const int l = (g / H_N) % S_LEN;
    const int n = g / (H_N * S_LEN);

    const float m = amask[(size_t)n * S_LEN + l];
    const float* pw = pos_weight + (size_t)l * HIDDEN + (size_t)h * D_DIM;
    const float* ps = psum + ((size_t)n * H_N + h) * D_DIM;

    float acc = 0.0f;
    #pragma unroll 8
    for (int d = 0; d < D_DIM; ++d)
        acc += elu1(pw[d]) * m * ps[d];
    z[g] = acc + EPS_F;
}

// ---------------------------------------------------------------------------
extern "C" void kernel_launch(void* const* d_in, const int* in_sizes, int n_in,
                              void* d_out, int out_size, void* d_ws, size_t ws_size,
                              hipStream_t stream) {
    // setup_inputs() order: q, v, attention_mask, head_mask, pos_weight
    const float* q  = (const float*)d_in[0];  (void)q;  // unused by reference
    const float* v  = (const float*)d_in[1];
    const float* am = (const float*)d_in[2];
    const float* hm = (const float*)d_in[3];
    const float* pw = (const float*)d_in[4];
    (void)in_sizes; (void)n_in; (void)out_size; (void)ws_size;

    float* out = (float*)d_out;
    float* pv   = (float*)d_ws;                                     // N*H*D*D floats (393 KB)
    float* psum = pv + (size_t)N_B * H_N * D_DIM * D_DIM;           // N*H*D floats (6 KB)

    // Stage 1: fp32 WMMA GEMM -> pv (fully overwritten, no init needed)
    pv_wmma_kernel<<<N_B * H_N * 16, 32, 0, stream>>>(v, pw, am, hm, pv);
    // Stage 2: column sums of p
    psum_kernel<<<N_B * H_N, D_DIM, 0, stream>>>(pw, am, psum);
    // Stage 3: bandwidth-bound broadcast outer product (dominant, ~403 MB stores)
    ppv_kernel<<<N_B * S_LEN * H_N, 256, 0, stream>>>(pw, am, pv, out);
    // Stage 4: normalizer appended after ppv in d_out
    const size_t PPV_COUNT = (size_t)N_B * S_LEN * H_N * D_DIM * D_DIM;
    z_kernel<<<(N_B * S_LEN * H_N + 255) / 256, 256, 0, stream>>>(
        pw, am, psum, out + PPV_COUNT);
}